// MyModel_75402445848582
// MI455X (gfx1250) — compile-verified
//
#include <hip/hip_runtime.h>
#include <math.h>

#define BATCH 256
#define CIN_  4
#define LEN   8192
#define NF_   8
#define UC_   16
#define EPSV  1e-5f

typedef float v2f __attribute__((ext_vector_type(2)));
typedef float v8f __attribute__((ext_vector_type(8)));

// D = A(16x4) * B(4x16) + C, fp32 WMMA (CDNA5)
__device__ __forceinline__ v8f wmma4(v2f a, v2f b, v8f c) {
  return __builtin_amdgcn_wmma_f32_16x16x4_f32(false, a, false, b, (short)0, c,
                                               false, false);
}

// 24 distinct fea0 offsets used by the fused conv1 (off in [-7,8]) and
// dilated conv2 (even off in [-14,16]) output stage.
__host__ __device__ __forceinline__ constexpr int off_of(int o) {
  return (o < 4) ? (-14 + 2 * o) : (o < 20) ? (o - 11) : (2 * o - 30);
}

__global__ void k_zero(float* __restrict__ p, int n) {
  int i = blockIdx.x * blockDim.x + threadIdx.x;
  if (i < n) p[i] = 0.f;
}

// ---------------------------------------------------------------------------
// Kernel 1: Gram matrix G[b] = c0 * c0^T, c0 recomputed on the fly.
// ---------------------------------------------------------------------------
#define T1 2048
__global__ __launch_bounds__(256) void k_gram(
    const float* __restrict__ x, const float* __restrict__ wc1,
    const float* __restrict__ g1, const float* __restrict__ bt1,
    const float* __restrict__ m1, const float* __restrict__ v1,
    float* __restrict__ G) {
  __shared__ float xs[CIN_][T1];
  __shared__ float gsh[256];
  __shared__ float wc1s[UC_][CIN_];
  __shared__ float sc1[UC_], sh1[UC_];

  int b = blockIdx.y;
  int l0 = blockIdx.x * T1;
  int tid = threadIdx.x;

  if (tid < UC_) {
    float inv = g1[tid] * rsqrtf(v1[tid] + EPSV);
    sc1[tid] = inv;
    sh1[tid] = bt1[tid] - m1[tid] * inv;
  }
  if (tid < UC_ * CIN_) wc1s[tid / CIN_][tid % CIN_] = wc1[tid];
  gsh[tid] = 0.f;
  for (int i = tid; i < CIN_ * T1; i += 256) {
    int ci = i / T1, c = i % T1;
    xs[ci][c] = x[((size_t)b * CIN_ + ci) * LEN + l0 + c];
  }
  __syncthreads();

  int wave = tid >> 5, lane = tid & 31;
  int ch = lane & 15;
  int base = wave * 256;
  float wr0 = wc1s[ch][0], wr1 = wc1s[ch][1], wr2 = wc1s[ch][2],
        wr3 = wc1s[ch][3];
  float sc = sc1[ch], sh = sh1[ch];
  v8f acc = {};
  for (int q = 0; q < 64; ++q) {
    int p = base + q * 4 + ((lane < 16) ? 0 : 2);
    float t0 = wr0 * xs[0][p] + wr1 * xs[1][p] + wr2 * xs[2][p] + wr3 * xs[3][p];
    float t1 = wr0 * xs[0][p + 1] + wr1 * xs[1][p + 1] + wr2 * xs[2][p + 1] +
               wr3 * xs[3][p + 1];
    v2f a;
    a.x = fmaxf(t0 * sc + sh, 0.f);
    a.y = fmaxf(t1 * sc + sh, 0.f);
    // For G = c0 * c0^T the A and B register images coincide.
    acc = wmma4(a, a, acc);
  }
  int N = lane & 15;
  int mh = (lane < 16) ? 0 : 8;
#pragma unroll
  for (int v = 0; v < 8; ++v) atomicAdd(&gsh[(v + mh) * 16 + N], acc[v]);
  __syncthreads();
  atomicAdd(&G[b * 256 + tid], gsh[tid]);
}

// ---------------------------------------------------------------------------
// Kernel 2: attn[b] = softmax(rowmax(G) - G) over last axis.
// ---------------------------------------------------------------------------
__global__ __launch_bounds__(32) void k_softmax(const float* __restrict__ G,
                                                float* __restrict__ attn) {
  int b = blockIdx.x;
  int lane = threadIdx.x;
  if (lane < 16) {
    float r[16];
    float t = -INFINITY;
#pragma unroll
    for (int d = 0; d < 16; ++d) {
      r[d] = G[b * 256 + lane * 16 + d];
      t = fmaxf(t, r[d]);
    }
    float m = -INFINITY;
#pragma unroll
    for (int d = 0; d < 16; ++d) {
      r[d] = t - r[d];
      m = fmaxf(m, r[d]);
    }
    float s = 0.f;
#pragma unroll
    for (int d = 0; d < 16; ++d) {
      r[d] = expf(r[d] - m);
      s += r[d];
    }
    float inv = 1.f / s;
#pragma unroll
    for (int d = 0; d < 16; ++d) attn[b * 256 + lane * 16 + d] = r[d] * inv;
  }
}

// ---------------------------------------------------------------------------
// Kernel 3: fully fused output stage, all GEMM-shaped work on WMMA.
// ---------------------------------------------------------------------------
#define LT 512
#define XH (LT + 64)  // x cols, global range [l0-32, l0+LT+32)
#define FH (LT + 32)  // fea0 cols, global j range [l0-16, l0+LT+16); 34 tiles

__global__ __launch_bounds__(256) void k_main(
    const float* __restrict__ x, const float* __restrict__ w00,
    const float* __restrict__ b00, const float* __restrict__ w01,
    const float* __restrict__ b01, const float* __restrict__ w02,
    const float* __restrict__ b02, const float* __restrict__ wc1,
    const float* __restrict__ g1, const float* __restrict__ bt1,
    const float* __restrict__ m1, const float* __restrict__ v1,
    const float* __restrict__ beta_cam, const float* __restrict__ wc2,
    const float* __restrict__ g2, const float* __restrict__ bt2,
    const float* __restrict__ m2, const float* __restrict__ v2,
    const float* __restrict__ attn, float* __restrict__ out) {
  __shared__ float xs[CIN_][XH];
  __shared__ float fs[NF_][FH];
  __shared__ float Wf[16][192];       // fused conv1/conv2 weights
  __shared__ float w00t[128][16];     // conv0 weights, K-major, N zero-padded
  __shared__ float attns[16][16];
  __shared__ float wc2s[16][16];
  __shared__ float wc1s[16][CIN_];
  __shared__ float sc1[16], sh1[16], sc2[16], sh2[16];
  __shared__ float b00s[NF_], b01s[NF_], b02s[NF_];

  int b = blockIdx.y;
  int l0 = blockIdx.x * LT;
  int tid = threadIdx.x;

  // ---- parameter staging -------------------------------------------------
  if (tid < 16) {
    float i1 = g1[tid] * rsqrtf(v1[tid] + EPSV);
    sc1[tid] = i1;
    sh1[tid] = bt1[tid] - m1[tid] * i1;
    float i2 = g2[tid] * rsqrtf(v2[tid] + EPSV);
    sc2[tid] = i2;
    sh2[tid] = bt2[tid] - m2[tid] * i2;
  }
  if (tid < 64) wc1s[tid / 4][tid % 4] = wc1[tid];
  if (tid >= 64 && tid < 64 + NF_) {
    int n = tid - 64;
    b00s[n] = b00[n];
    b01s[n] = b01[n];
    b02s[n] = b02[n];
  }
  if (tid < 256) {
    attns[tid / 16][tid % 16] = attn[b * 256 + tid];
    wc2s[tid / 16][tid % 16] = wc2[tid];
  }
  // conv0 weights transposed to B layout: w00t[k = ci*32+t][n = co]
  for (int i = tid; i < 128 * 16; i += 256) {
    int k = i / 16, n = i % 16;
    int ci = k >> 5, t = k & 31;
    w00t[k][n] = (n < NF_) ? w00[n * 128 + ci * 32 + t] : 0.f;
  }
  // Combined conv1/conv2 weights, K = (offset o)*8 + fea0-channel ci.
  for (int i = tid; i < 16 * 192; i += 256) {
    int n = i / 192, k = i % 192;
    int o = k >> 3, ci = k & 7;
    int off = off_of(o);
    float wv = 0.f;
    if (n < 8) {
      int kk = off + 7;
      if (kk >= 0 && kk < 16) wv = w01[n * 128 + ci * 16 + kk];
    } else if (((off + 14) & 1) == 0) {
      int kk = (off + 14) >> 1;
      if (kk >= 0 && kk < 16) wv = w02[(n - 8) * 128 + ci * 16 + kk];
    }
    Wf[n][k] = wv;
  }
  // x tile with halo (zero-filled out of range)
  for (int i = tid; i < CIN_ * XH; i += 256) {
    int ci = i / XH, c = i % XH;
    int gx = l0 + c - 32;
    xs[ci][c] = (gx >= 0 && gx < LEN)
                    ? x[((size_t)b * CIN_ + ci) * LEN + gx]
                    : 0.f;
  }
  __syncthreads();

  int wave = tid >> 5, lane = tid & 31;
  int hh = (lane < 16) ? 0 : 1;  // lane half
  int row = lane & 15;

  // ---- fea0 = conv0(x) + b00 via WMMA ------------------------------------
  // D(16 j-pos x 16 co) = Xgather(16 x 128) * w00t(128 x 16), K = (ci, t)
  for (int s = wave; s < FH / 16; s += 8) {
    int j0r = s * 16;
    v8f da = {}, db = {};
#pragma unroll
    for (int kk = 0; kk < 32; ++kk) {
      int ci = kk >> 3;
      int tb = (4 * kk) & 31;
      int kr = 4 * kk + 2 * hh;
      v2f a, bv;
      int xc = j0r + row + tb + 2 * hh;
      a.x = xs[ci][xc];
      a.y = xs[ci][xc + 1];
      bv.x = w00t[kr][row];
      bv.y = w00t[kr + 1][row];
      if (kk & 1) db = wmma4(a, bv, db);
      else        da = wmma4(a, bv, da);
    }
    // D layout: lane half hh holds j-positions j0r + v + 8*hh, co = row.
    if (row < NF_) {
      float bias = b00s[row];
#pragma unroll
      for (int v = 0; v < 8; ++v) {
        int jr = j0r + v + 8 * hh;
        int j = l0 + jr - 16;
        fs[row][jr] = (j >= 0 && j <= LEN) ? (da[v] + db[v] + bias) : 0.f;
      }
    }
  }
  __syncthreads();

  // ---- per-wave 16x16 output tiles ---------------------------------------
  float beta = beta_cam[0];

  for (int s = wave; s < LT / 16; s += 8) {
    int p = s * 16;
    int colr = p + row;
    int gcol = l0 + colr;

    // c0 = bn_relu(wc1 x x) via one WMMA (K = 4 input channels)
    v8f dc = {};
    {
      v2f a, bv;
      a.x = wc1s[row][2 * hh];
      a.y = wc1s[row][2 * hh + 1];
      bv.x = xs[2 * hh][colr + 32];
      bv.y = xs[2 * hh + 1][colr + 32];
      dc = wmma4(a, bv, dc);
    }
    float c0d[8];
#pragma unroll
    for (int v = 0; v < 8; ++v) {
      int ch = v + hh * 8;
      c0d[v] = fmaxf(dc[v] * sc1[ch] + sh1[ch], 0.f);
    }
    // full 16-channel column via xor-16 shuffle (same column both halves)
    float c0full[16];
#pragma unroll
    for (int v = 0; v < 8; ++v) {
      float o = __shfl_xor(c0d[v], 16, 32);
      c0full[v + hh * 8] = c0d[v];
      c0full[v + (1 - hh) * 8] = o;
    }

    // feat = attn (16x16) x c0 (16x16)
    v8f d1 = {};
#pragma unroll
    for (int kk = 0; kk < 4; ++kk) {
      int kb = 4 * kk + 2 * hh;
      v2f a, bv;
      a.x = attns[row][kb];
      a.y = attns[row][kb + 1];
      bv.x = c0full[kb];
      bv.y = c0full[kb + 1];
      d1 = wmma4(a, bv, d1);
    }

    // c1 = beta*feat + c0 (D layout), complete columns via xor-16
    float c1d[8];
#pragma unroll
    for (int v = 0; v < 8; ++v) c1d[v] = beta * d1[v] + c0d[v];
    float c1full[16];
#pragma unroll
    for (int v = 0; v < 8; ++v) {
      float o = __shfl_xor(c1d[v], 16, 32);
      c1full[v + hh * 8] = c1d[v];
      c1full[v + (1 - hh) * 8] = o;
    }

    // conv2 = wc2 (16x16) x c1 (16x16)
    v8f d2 = {};
#pragma unroll
    for (int kk = 0; kk < 4; ++kk) {
      int kb = 4 * kk + 2 * hh;
      v2f a, bv;
      a.x = wc2s[row][kb];
      a.y = wc2s[row][kb + 1];
      bv.x = c1full[kb];
      bv.y = c1full[kb + 1];
      d2 = wmma4(a, bv, d2);
    }

    // conv_fusion = Wf (16x192) x fea0 gather (192x16), dual accumulators
    v8f d3a = {}, d3b = {};
#pragma unroll
    for (int kk = 0; kk < 48; ++kk) {
      int o = kk >> 1;
      int off = off_of(o);
      int cib = ((kk & 1) << 2) + 2 * hh;
      int kidx = 4 * kk + 2 * hh;
      v2f a, bv;
      a.x = Wf[row][kidx];
      a.y = Wf[row][kidx + 1];
      int fi = colr + off + 16;
      bv.x = fs[cib][fi];
      bv.y = fs[cib + 1][fi];
      if (kk & 1) d3b = wmma4(a, bv, d3b);
      else        d3a = wmma4(a, bv, d3a);
    }

    // epilogue: BN2+ReLU, conv1/conv2 biases, fea2 pad zeros, store
    bool fea2ok = (gcol >= 1) && (gcol < LEN - 2);
#pragma unroll
    for (int v = 0; v < 8; ++v) {
      int ch = v + hh * 8;
      float c2 = fmaxf(d2[v] * sc2[ch] + sh2[ch], 0.f);
      float f3 = d3a[v] + d3b[v];
      float fu = (hh == 0) ? (f3 + b01s[v])
                           : (fea2ok ? (f3 + b02s[v]) : 0.f);
      out[((size_t)b * 16 + ch) * LEN + gcol] = fu + c2;
    }
  }
}

// ---------------------------------------------------------------------------
extern "C" void kernel_launch(void* const* d_in, const int* in_sizes, int n_in,
                              void* d_out, int out_size, void* d_ws,
                              size_t ws_size, hipStream_t stream) {
  (void)in_sizes; (void)n_in; (void)out_size; (void)ws_size;
  const float* x    = (const float*)d_in[0];
  const float* w00  = (const float*)d_in[1];
  const float* b00  = (const float*)d_in[2];
  const float* w01  = (const float*)d_in[3];
  const float* b01  = (const float*)d_in[4];
  const float* w02  = (const float*)d_in[5];
  const float* b02  = (const float*)d_in[6];
  const float* wc1  = (const float*)d_in[7];
  const float* g1   = (const float*)d_in[8];
  const float* bt1  = (const float*)d_in[9];
  const float* m1   = (const float*)d_in[10];
  const float* v1   = (const float*)d_in[11];
  const float* beta = (const float*)d_in[12];
  const float* wc2  = (const float*)d_in[13];
  const float* g2   = (const float*)d_in[14];
  const float* bt2  = (const float*)d_in[15];
  const float* m2   = (const float*)d_in[16];
  const float* v2   = (const float*)d_in[17];
  float* out = (float*)d_out;

  float* G    = (float*)d_ws;          // BATCH * 256 floats
  float* attn = G + BATCH * 256;       // BATCH * 256 floats

  k_zero<<<(BATCH * 256 + 255) / 256, 256, 0, stream>>>(G, BATCH * 256);

  dim3 gg(LEN / T1, BATCH);
  k_gram<<<gg, 256, 0, stream>>>(x, wc1, g1, bt1, m1, v1, G);

  k_softmax<<<BATCH, 32, 0, stream>>>(G, attn);

  dim3 gm(LEN / LT, BATCH);
  k_main<<<gm, 256, 0, stream>>>(x, w00, b00, w01, b01, w02, b02, wc1, g1, bt1,
                                 m1, v1, beta, wc2, g2, bt2, m2, v2, attn, out);
}